// PillarFeatureNet_60705067762261
// MI455X (gfx1250) — compile-verified
//
#include <hip/hip_runtime.h>
#include <hip/hip_bf16.h>

// ---------------------------------------------------------------------------
// PillarFeatureNet for MI455X (gfx1250, wave32).
//  Pass 1: per-pillar stats -> global sum(feats[9]) and sum(feats x feats)[45]
//  Pass 2: fold BN batch stats analytically (x = feats@W is linear), fuse
//          gamma/beta into per-channel (a,b); pad W to [12,64].
//  Pass 3: per-pillar GEMM with V_WMMA_F32_16X16X4_F32 (16x16 tiles, 3
//          k-steps of 4), fused affine+ReLU+max-over-points.
// ---------------------------------------------------------------------------

typedef float v2f __attribute__((ext_vector_type(2)));
typedef float v8f __attribute__((ext_vector_type(8)));

#define PPB    8      // pillars (waves) per 256-thread block
#define P_PTS  100
#define C_IN   4
#define TILES  7      // ceil(100/16)

// ws layout (floats): [0..8] sum f, [9..53] upper-tri sum f_i f_j,
// [64..127] a (scale), [128..191] b (bias), [192..959] padded W[12][64]
#define WS_A   64
#define WS_B   128
#define WS_WP  192

#define VXc    0.2f
#define XOFFc  0.1f      // VX/2 + 0.0
#define YOFFc  (-39.9f)  // VY/2 - 40.0

__device__ __forceinline__ float wave_sum32(float v) {
#pragma unroll
  for (int off = 16; off > 0; off >>= 1) v += __shfl_xor(v, off, 32);
  return v;
}

__global__ void pfn_zero(float* __restrict__ ws) {
  if (threadIdx.x < 54) ws[threadIdx.x] = 0.0f;
}

// ------------------------- Pass 1: moment statistics ------------------------
__global__ void __launch_bounds__(32 * PPB)
pfn_stats(const float* __restrict__ feat, const int* __restrict__ nvox,
          const int* __restrict__ coors, float* __restrict__ ws, int K) {
  const int lane = threadIdx.x & 31;
  const int k    = blockIdx.x * PPB + (threadIdx.x >> 5);
  __shared__ float red[54];
  if (threadIdx.x < 54) red[threadIdx.x] = 0.0f;
  __syncthreads();

  if (k < K) {
    const int n = nvox[k];
    const float cx = (float)coors[k * 4 + 3] * VXc + XOFFc;
    const float cy = (float)coors[k * 4 + 2] * VXc + YOFFc;

    // pillar mean: sum xyz over ALL P points, divide by n (matches reference)
    float sx = 0.f, sy = 0.f, sz = 0.f;
    for (int p = lane; p < P_PTS; p += 32) {
      const float4 v = *(const float4*)(feat + ((size_t)k * P_PTS + p) * C_IN);
      sx += v.x; sy += v.y; sz += v.z;
    }
    sx = wave_sum32(sx); sy = wave_sum32(sy); sz = wave_sum32(sz);
    const float invn = 1.0f / (float)n;
    const float mx = sx * invn, my = sy * invn, mz = sz * invn;

    float s[9];  float q[45];
#pragma unroll
    for (int i = 0; i < 9; i++)  s[i] = 0.f;
#pragma unroll
    for (int i = 0; i < 45; i++) q[i] = 0.f;

    // only valid (unmasked) points contribute non-zero feats
    for (int p = lane; p < n; p += 32) {
      const float4 v = *(const float4*)(feat + ((size_t)k * P_PTS + p) * C_IN);
      const float f[9] = {v.x, v.y, v.z, v.w,
                          v.x - mx, v.y - my, v.z - mz,
                          v.x - cx, v.y - cy};
      int qi = 0;
#pragma unroll
      for (int i = 0; i < 9; i++) {
        s[i] += f[i];
#pragma unroll
        for (int j = i; j < 9; j++) q[qi++] += f[i] * f[j];
      }
    }
#pragma unroll
    for (int i = 0; i < 9; i++)  s[i] = wave_sum32(s[i]);
#pragma unroll
    for (int i = 0; i < 45; i++) q[i] = wave_sum32(q[i]);

    if (lane == 0) {
#pragma unroll
      for (int i = 0; i < 9; i++)  atomicAdd(&red[i], s[i]);
#pragma unroll
      for (int i = 0; i < 45; i++) atomicAdd(&red[9 + i], q[i]);
    }
  }
  __syncthreads();
  if (threadIdx.x < 54) atomicAdd(&ws[threadIdx.x], red[threadIdx.x]);
}

// ---------------- Pass 2: fold BN stats, fuse affine, pad W -----------------
__device__ __forceinline__ int triidx(int i, int j) {  // i <= j, 9x9 upper tri
  return i * 9 - (i * (i - 1)) / 2 + (j - i);
}

__global__ void pfn_finalize(const float* __restrict__ W,
                             const float* __restrict__ gamma,
                             const float* __restrict__ beta,
                             float* __restrict__ ws, int K) {
  const int u = threadIdx.x;  // 64 threads, one output channel each
  const float invN = 1.0f / ((float)K * (float)P_PTS);
  float wcol[9];
#pragma unroll
  for (int i = 0; i < 9; i++) wcol[i] = W[i * 64 + u];

  float mu = 0.f;
#pragma unroll
  for (int i = 0; i < 9; i++) mu += ws[i] * wcol[i];
  mu *= invN;

  float e2 = 0.f;
#pragma unroll
  for (int i = 0; i < 9; i++) {
#pragma unroll
    for (int j = 0; j < 9; j++) {
      const int a = i < j ? i : j;
      const int b = i < j ? j : i;
      e2 += ws[9 + triidx(a, b)] * wcol[i] * wcol[j];
    }
  }
  e2 *= invN;

  const float var = e2 - mu * mu;
  const float inv = rsqrtf(var + 1e-3f);
  const float a_  = gamma[u] * inv;
  const float b_  = beta[u] - mu * a_;
  ws[WS_A + u] = a_;
  ws[WS_B + u] = b_;
#pragma unroll
  for (int r = 0; r < 12; r++)
    ws[WS_WP + r * 64 + u] = (r < 9) ? W[r * 64 + u] : 0.0f;
}

// -------------- Pass 3: WMMA GEMM + fused BN/ReLU/max-over-P ----------------
__global__ void __launch_bounds__(32 * PPB)
pfn_main(const float* __restrict__ feat, const int* __restrict__ nvox,
         const int* __restrict__ coors, const float* __restrict__ ws,
         float* __restrict__ out, int K) {
  const int lane = threadIdx.x & 31;
  const int k    = blockIdx.x * PPB + (threadIdx.x >> 5);
  if (k >= K) return;              // wave-uniform; EXEC stays all-1s for WMMA

  const int nIdx = lane & 15;      // N column within 16-wide tile
  const int hi   = lane >> 4;      // lane half selects K pair {0,1} vs {2,3}

  // per-lane fused affine for the 4 N-tiles
  float a_s[4], b_s[4];
#pragma unroll
  for (int nt = 0; nt < 4; nt++) {
    a_s[nt] = ws[WS_A + nt * 16 + nIdx];
    b_s[nt] = ws[WS_B + nt * 16 + nIdx];
  }

  // B fragments: B[4x16] per k-step s, n-tile nt.
  // VGPR0 = rows {K=s*4+0 | s*4+2}, VGPR1 = rows {K=s*4+1 | s*4+3} by lane half.
  const float* wp = ws + WS_WP;
  v2f bf[3][4];
#pragma unroll
  for (int s = 0; s < 3; s++) {
    const int kb = s * 4 + hi * 2;
#pragma unroll
    for (int nt = 0; nt < 4; nt++) {
      v2f t;
      t.x = wp[kb * 64 + nt * 16 + nIdx];
      t.y = wp[(kb + 1) * 64 + nt * 16 + nIdx];
      bf[s][nt] = t;
    }
  }

  const int n = nvox[k];
  const float cx = (float)coors[k * 4 + 3] * VXc + XOFFc;
  const float cy = (float)coors[k * 4 + 2] * VXc + YOFFc;

  float sx = 0.f, sy = 0.f, sz = 0.f;
  for (int p = lane; p < P_PTS; p += 32) {
    const float4 v = *(const float4*)(feat + ((size_t)k * P_PTS + p) * C_IN);
    sx += v.x; sy += v.y; sz += v.z;
  }
  sx = wave_sum32(sx); sy = wave_sum32(sy); sz = wave_sum32(sz);
  const float invn = 1.0f / (float)n;
  const float mx = sx * invn, my = sy * invn, mz = sz * invn;

  v8f runmax[4];
#pragma unroll
  for (int nt = 0; nt < 4; nt++)
    runmax[nt] = (v8f){0.f, 0.f, 0.f, 0.f, 0.f, 0.f, 0.f, 0.f};

  for (int t = 0; t < TILES; t++) {
    const int p = t * 16 + nIdx;   // my A-matrix row (point index)
    float x = 0.f, y = 0.f, z = 0.f, w = 0.f;
    if (p < P_PTS) {
      const float4 v = *(const float4*)(feat + ((size_t)k * P_PTS + p) * C_IN);
      x = v.x; y = v.y; z = v.z; w = v.w;
    }
    const float m = (p < n) ? 1.0f : 0.0f;
    float f[12];
    f[0] = x * m;  f[1] = y * m;  f[2] = z * m;  f[3] = w * m;
    f[4] = (x - mx) * m; f[5] = (y - my) * m; f[6] = (z - mz) * m;
    f[7] = (x - cx) * m; f[8] = (y - cy) * m;
    f[9] = 0.f; f[10] = 0.f; f[11] = 0.f;

    v8f c[4];
#pragma unroll
    for (int nt = 0; nt < 4; nt++)
      c[nt] = (v8f){0.f, 0.f, 0.f, 0.f, 0.f, 0.f, 0.f, 0.f};

#pragma unroll
    for (int s = 0; s < 3; s++) {
      v2f af;
      af.x = hi ? f[s * 4 + 2] : f[s * 4 + 0];
      af.y = hi ? f[s * 4 + 3] : f[s * 4 + 1];
#pragma unroll
      for (int nt = 0; nt < 4; nt++) {
        c[nt] = __builtin_amdgcn_wmma_f32_16x16x4_f32(
            false, af, false, bf[s][nt], (short)0, c[nt], false, false);
      }
    }

    // fused BN affine + ReLU; pad rows p>=100 contribute 0 (safe: relu>=0)
#pragma unroll
    for (int nt = 0; nt < 4; nt++) {
#pragma unroll
      for (int j = 0; j < 8; j++) {
        const int prow = t * 16 + j + hi * 8;
        float yv = c[nt][j] * a_s[nt] + b_s[nt];
        yv = fmaxf(yv, 0.0f);
        yv = (prow < P_PTS) ? yv : 0.0f;
        runmax[nt][j] = fmaxf(runmax[nt][j], yv);
      }
    }
  }

#pragma unroll
  for (int nt = 0; nt < 4; nt++) {
    float mv = runmax[nt][0];
#pragma unroll
    for (int j = 1; j < 8; j++) mv = fmaxf(mv, runmax[nt][j]);
    mv = fmaxf(mv, __shfl_xor(mv, 16, 32));  // merge M=0..7 with M=8..15 halves
    if (lane < 16) out[(size_t)k * 64 + nt * 16 + lane] = mv;
  }
}

// ---------------------------------------------------------------------------
extern "C" void kernel_launch(void* const* d_in, const int* in_sizes, int n_in,
                              void* d_out, int out_size, void* d_ws,
                              size_t ws_size, hipStream_t stream) {
  const float* feat  = (const float*)d_in[0];
  const int*   nvox  = (const int*)d_in[1];
  const int*   coors = (const int*)d_in[2];
  const float* W     = (const float*)d_in[3];
  const float* gamma = (const float*)d_in[4];
  const float* beta  = (const float*)d_in[5];
  float* out = (float*)d_out;
  float* ws  = (float*)d_ws;

  const int K = in_sizes[1];                 // num_voxels count = #pillars
  const int blocks = (K + PPB - 1) / PPB;

  pfn_zero<<<1, 64, 0, stream>>>(ws);
  pfn_stats<<<blocks, 32 * PPB, 0, stream>>>(feat, nvox, coors, ws, K);
  pfn_finalize<<<1, 64, 0, stream>>>(W, gamma, beta, ws, K);
  pfn_main<<<blocks, 32 * PPB, 0, stream>>>(feat, nvox, coors, ws, out, K);
}